// D3BJ_NB_47991964566172
// MI455X (gfx1250) — compile-verified
//
#include <hip/hip_runtime.h>

#ifndef A1C
#define A1C 0.5299f
#define A2C 4.6f
#define S6C 1.0f
#define S8C 2.0f
#define ANG2BOHR 1.889716f
#define HARTREE2EV 27.211368
#define EPSC 1e-6f
#endif

// ---- CDNA5 async global->LDS availability probes -------------------------
#if defined(__has_builtin)
#  if __has_builtin(__builtin_amdgcn_global_load_async_to_lds_b128)
#    define HAVE_ASYNC_B128 1
#  endif
#  if __has_builtin(__builtin_amdgcn_global_load_async_to_lds_b32)
#    define HAVE_ASYNC_B32 1
#  endif
#  if __has_builtin(__builtin_amdgcn_s_wait_asynccnt)
#    define HAVE_WAIT_ASYNC 1
#  endif
#endif

typedef int v4i __attribute__((ext_vector_type(4)));
typedef __attribute__((address_space(1))) int GInt;   // global scalar (AS1)
typedef __attribute__((address_space(3))) int LInt;   // LDS scalar (AS3)
typedef __attribute__((address_space(1))) v4i GV4i;   // global v4i (AS1)
typedef __attribute__((address_space(3))) v4i LV4i;   // LDS v4i (AS3)

static __device__ __forceinline__ void wait_async0() {
#if defined(HAVE_WAIT_ASYNC)
    __builtin_amdgcn_s_wait_asynccnt(0);
#else
    asm volatile("s_wait_asynccnt 0" ::: "memory");
#endif
    asm volatile("" ::: "memory");  // keep LDS reads after the drain
}

static __device__ __forceinline__ float pair_energy(float4 Ai, float2 Bi,
                                                    float4 Aj, float2 Bj,
                                                    unsigned char padded) {
    const float dx = Aj.x - Ai.x;
    const float dy = Aj.y - Ai.y;
    const float dz = Aj.z - Ai.z;
    float d2 = dx * dx + dy * dy + dz * dz;
    if (padded) d2 = 1.0f;

    const float d2b = d2 * (ANG2BOHR * ANG2BOHR);  // d_ij^2 in bohr^2
    const float d6 = d2b * d2b * d2b;
    const float d8 = d6 * d2b;

    const float c6i = Ai.w, c6j = Aj.w;
    const float ai = Bi.x, aj = Bj.x;
    const float denom = fmaxf(c6i * aj / ai + c6j * ai / aj, EPSC);
    const float c6ij = 2.0f * c6i * c6j / denom;

    const float rrij = 3.0f * Bi.y * Bj.y;
    const float r0 = A1C * sqrtf(rrij) + A2C;
    const float r02 = r0 * r0;
    const float r06 = r02 * r02 * r02;
    const float r08 = r06 * r02;

    return c6ij * (S6C / (d6 + r06) + S8C * rrij / (d8 + r08));
}

// ---- hot kernel (defined FIRST so the disasm snippet shows it) -----------
__global__ void __launch_bounds__(256)
d3bj_pairs_packed(const float4* __restrict__ atomA,
                  const float2* __restrict__ atomB,
                  const int* __restrict__ idx_j,
                  const unsigned char* __restrict__ pad_mask,
                  double* __restrict__ acc,
                  int n_pairs, int n_nb, int nb_shift) {
    __shared__ int s_idx[256];
    __shared__ float s_wsum[8];

    const int base = blockIdx.x * 256;
    const int p = base + (int)threadIdx.x;
    const bool valid = (p < n_pairs);
    const bool full = (base + 256) <= n_pairs;  // block-uniform

    // ---- stage this block's idx_j tile via CDNA5 async global->LDS -------
#if defined(HAVE_ASYNC_B128)
    if (full && threadIdx.x < 64) {
        v4i* g = (v4i*)(idx_j + base + threadIdx.x * 4);
        v4i* l = (v4i*)&s_idx[threadIdx.x * 4];
        __builtin_amdgcn_global_load_async_to_lds_b128(
            (GV4i*)g, (LV4i*)l, 0, 0);
    }
#elif defined(HAVE_ASYNC_B32)
    if (full) {
        int* g = (int*)(idx_j + p);
        __builtin_amdgcn_global_load_async_to_lds_b32(
            (GInt*)g, (LInt*)&s_idx[threadIdx.x], 0, 0);
    }
#endif

    // ---- overlap: i-side loads + mask while the async engine runs --------
    float4 Ai = make_float4(0.f, 0.f, 0.f, 0.f);
    float2 Bi = make_float2(1.f, 0.f);
    unsigned char m = 0;
    if (valid) {
        const int i = (nb_shift >= 0) ? (p >> nb_shift) : (p / n_nb);
        Ai = atomA[i];
        Bi = atomB[i];
        m = pad_mask[p];
    }

    // ---- drain async + pick up j ----------------------------------------
    int j = 0;
#if defined(HAVE_ASYNC_B128)
    if (full) {
        wait_async0();
        __syncthreads();
        j = s_idx[threadIdx.x];
    } else if (valid) {
        j = idx_j[p];
    }
#elif defined(HAVE_ASYNC_B32)
    if (full) {
        wait_async0();
        j = s_idx[threadIdx.x];   // own element only; wave-local drain is enough
    } else if (valid) {
        j = idx_j[p];
    }
#else
    if (valid) j = idx_j[p];
#endif

    float e = 0.0f;
    if (valid) {
        const float4 Aj = atomA[j];
        const float2 Bj = atomB[j];
        e = pair_energy(Ai, Bi, Aj, Bj, m);
    }

    // ---- wave32 shuffle tree + block combine -----------------------------
    for (int off = 16; off > 0; off >>= 1)
        e += __shfl_down(e, off, 32);

    const int lane = threadIdx.x & 31;
    const int wid = threadIdx.x >> 5;
    if (lane == 0) s_wsum[wid] = e;
    __syncthreads();

    if (wid == 0) {
        float v = (lane < 8) ? s_wsum[lane] : 0.0f;
        for (int off = 4; off > 0; off >>= 1)
            v += __shfl_down(v, off, 32);
        if (lane == 0) atomicAdd(acc, (double)v);  // global_atomic_add_f64
    }
}

// ---- fallback kernel (unpacked inputs; used only if ws too small) --------
__global__ void __launch_bounds__(256)
d3bj_pairs_raw(const float* __restrict__ coord,
               const float* __restrict__ c6,
               const float* __restrict__ alpha,
               const float* __restrict__ r4r2,
               const int* __restrict__ numbers,
               const int* __restrict__ idx_j,
               const unsigned char* __restrict__ pad_mask,
               double* __restrict__ acc,
               int n_pairs, int n_nb) {
    __shared__ float s_wsum[8];
    const int p = blockIdx.x * 256 + (int)threadIdx.x;

    float e = 0.0f;
    if (p < n_pairs) {
        const int i = p / n_nb;
        const int j = idx_j[p];
        const float4 Ai = make_float4(coord[3 * i], coord[3 * i + 1], coord[3 * i + 2], c6[i]);
        const float2 Bi = make_float2(fmaxf(alpha[i], EPSC), r4r2[numbers[i]]);
        const float4 Aj = make_float4(coord[3 * j], coord[3 * j + 1], coord[3 * j + 2], c6[j]);
        const float2 Bj = make_float2(fmaxf(alpha[j], EPSC), r4r2[numbers[j]]);
        e = pair_energy(Ai, Bi, Aj, Bj, pad_mask[p]);
    }

    for (int off = 16; off > 0; off >>= 1)
        e += __shfl_down(e, off, 32);

    const int lane = threadIdx.x & 31;
    const int wid = threadIdx.x >> 5;
    if (lane == 0) s_wsum[wid] = e;
    __syncthreads();

    if (wid == 0) {
        float v = (lane < 8) ? s_wsum[lane] : 0.0f;
        for (int off = 4; off > 0; off >>= 1)
            v += __shfl_down(v, off, 32);
        if (lane == 0) atomicAdd(acc, (double)v);
    }
}

__global__ void d3bj_init_acc(double* acc) {
    if (blockIdx.x == 0 && threadIdx.x == 0) *acc = 0.0;
}

// Pack per-atom gather payload: atomA = {x,y,z,c6}, atomB = {clip(alpha), r4r2[Z]}
__global__ void d3bj_pack_atoms(const float* __restrict__ coord,
                                const float* __restrict__ c6,
                                const float* __restrict__ alpha,
                                const float* __restrict__ r4r2,
                                const int* __restrict__ numbers,
                                float4* __restrict__ atomA,
                                float2* __restrict__ atomB,
                                int n_atoms) {
    int a = blockIdx.x * blockDim.x + threadIdx.x;
    if (a >= n_atoms) return;
    atomA[a] = make_float4(coord[3 * a + 0], coord[3 * a + 1], coord[3 * a + 2], c6[a]);
    float al = fmaxf(alpha[a], EPSC);
    atomB[a] = make_float2(al, r4r2[numbers[a]]);
}

__global__ void d3bj_finalize(const double* __restrict__ acc, float* __restrict__ out) {
    if (blockIdx.x == 0 && threadIdx.x == 0)
        out[0] = (float)(-0.5 * (double)HARTREE2EV * (*acc));
}

// ---- host launch ---------------------------------------------------------

extern "C" void kernel_launch(void* const* d_in, const int* in_sizes, int n_in,
                              void* d_out, int out_size, void* d_ws, size_t ws_size,
                              hipStream_t stream) {
    const float* coord = (const float*)d_in[0];
    const float* c6 = (const float*)d_in[1];
    const float* alpha = (const float*)d_in[2];
    const float* r4r2 = (const float*)d_in[3];
    const int* idx_j = (const int*)d_in[4];
    const int* numbers = (const int*)d_in[5];
    const unsigned char* pad_mask = (const unsigned char*)d_in[6];

    const int n_atoms = in_sizes[1];
    const int n_pairs = in_sizes[4];
    const int n_nb = n_pairs / n_atoms;

    // shift for power-of-two neighbor counts (n_nb == 64 in practice)
    int nb_shift = -1;
    if (n_nb > 0 && (n_nb & (n_nb - 1)) == 0) {
        nb_shift = 0;
        while ((1 << nb_shift) < n_nb) ++nb_shift;
    }

    double* acc = (double*)d_ws;
    char* ws_bytes = (char*)d_ws;
    float4* atomA = (float4*)(ws_bytes + 16);
    float2* atomB = (float2*)(ws_bytes + 16 + (size_t)n_atoms * sizeof(float4));
    const size_t need = 16 + (size_t)n_atoms * (sizeof(float4) + sizeof(float2));
    const bool packed = (ws_size >= need);

    d3bj_init_acc<<<1, 1, 0, stream>>>(acc);

    const int pair_blocks = (n_pairs + 255) / 256;
    if (packed) {
        d3bj_pack_atoms<<<(n_atoms + 255) / 256, 256, 0, stream>>>(
            coord, c6, alpha, r4r2, numbers, atomA, atomB, n_atoms);
        d3bj_pairs_packed<<<pair_blocks, 256, 0, stream>>>(
            atomA, atomB, idx_j, pad_mask, acc, n_pairs, n_nb, nb_shift);
    } else {
        d3bj_pairs_raw<<<pair_blocks, 256, 0, stream>>>(
            coord, c6, alpha, r4r2, numbers, idx_j, pad_mask, acc, n_pairs, n_nb);
    }

    d3bj_finalize<<<1, 1, 0, stream>>>(acc, (float*)d_out);
}